// Unconditional_LSTM_6914897346673
// MI455X (gfx1250) — compile-verified
//
#include <hip/hip_runtime.h>

typedef __attribute__((ext_vector_type(2))) float v2f;
typedef __attribute__((ext_vector_type(8))) float v8f;

#define HDIM 1024
#define NTILES_H 64 /* 1024/16 */

__device__ __forceinline__ float fast_sigmoid(float x) {
    return 1.0f / (1.0f + __expf(-x));
}
__device__ __forceinline__ float fast_tanh(float x) {
    // 2*sigmoid(2x)-1 ; correct saturation at +-1 for large |x|
    return 2.0f / (1.0f + __expf(-2.0f * x)) - 1.0f;
}

__global__ __launch_bounds__(256) void lstm_fused_wmma_kernel(
    const float* __restrict__ seq,   // (T,3)
    const float* __restrict__ Wih,   // (4096,3) row-major
    const float* __restrict__ bih,   // (4096)
    const float* __restrict__ bhh,   // (4096)
    const float* __restrict__ Wout,  // (7,1024) row-major
    const float* __restrict__ bout,  // (7)
    const float* __restrict__ zn,    // (T,2)
    const float* __restrict__ un,    // (T,1)
    float* __restrict__ out)         // (T,3)
{
    const int lane = threadIdx.x & 31;
    const int wave = threadIdx.x >> 5;
    const int tile = blockIdx.x * 8 + wave;   // 16 sequence rows per wave
    const int t0   = tile * 16;

    const int half = lane >> 4;   // 0: K={0,1} / rows 0-7 of C ; 1: K={2,3} / rows 8-15 of C
    const int l15  = lane & 15;
    const int kA   = half ? 2 : 0;

    // ---- A operand: 16x4 f32 tile of sequence, K=3 zero-padded to 4 ----
    // ISA layout (16x4 f32 A): lanes 0-15 M=0..15 {V0:K=0, V1:K=1}; lanes 16-31 {V0:K=2, V1:K=3}
    v2f a;
    a.x = seq[(t0 + l15) * 3 + kA];
    a.y = half ? 0.0f : seq[(t0 + l15) * 3 + 1];

    // partial sums of lin[row][k] over this lane's 64 columns
    float acc[8][7];
#pragma unroll
    for (int r = 0; r < 8; ++r)
#pragma unroll
        for (int k = 0; k < 7; ++k) acc[r][k] = 0.0f;

    for (int ht = 0; ht < NTILES_H; ++ht) {
        const int n = ht * 16 + l15;  // column within H handled by this lane

        // ---- B operands: 4x16 f32 slices of W_ih.T for gates i,g,o ----
        // B layout: V0 {lanes0-15:K=0, lanes16-31:K=2}, V1 {K=1, K=3(=0 pad)}
        const int ci = (0 * HDIM + n) * 3;
        const int cg = (2 * HDIM + n) * 3;
        const int co = (3 * HDIM + n) * 3;
        v2f bi, bg, bo;
        bi.x = Wih[ci + kA]; bi.y = half ? 0.0f : Wih[ci + 1];
        bg.x = Wih[cg + kA]; bg.y = half ? 0.0f : Wih[cg + 1];
        bo.x = Wih[co + kA]; bo.y = half ? 0.0f : Wih[co + 1];

        v8f cz = {};
        // D = A x B : 16x16 gate pre-activation tiles (full f32)
        v8f gi = __builtin_amdgcn_wmma_f32_16x16x4_f32(false, a, false, bi, (short)0, cz, false, false);
        v8f gg = __builtin_amdgcn_wmma_f32_16x16x4_f32(false, a, false, bg, (short)0, cz, false, false);
        v8f go = __builtin_amdgcn_wmma_f32_16x16x4_f32(false, a, false, bo, (short)0, cz, false, false);

        // biases depend only on column n (same for all 16 rows)
        const float biasi = bih[0 * HDIM + n] + bhh[0 * HDIM + n];
        const float biasg = bih[2 * HDIM + n] + bhh[2 * HDIM + n];
        const float biaso = bih[3 * HDIM + n] + bhh[3 * HDIM + n];

        // W_out column slice for this lane's column n (coalesced across l15)
        float w[7];
#pragma unroll
        for (int k = 0; k < 7; ++k) w[k] = Wout[k * HDIM + n];

        // activations + fold into output projection; C layout: VGPR r -> row (r + 8*half)
#pragma unroll
        for (int r = 0; r < 8; ++r) {
            const float iv = fast_sigmoid(gi[r] + biasi);
            const float gv = fast_tanh(gg[r] + biasg);
            const float ov = fast_sigmoid(go[r] + biaso);
            const float cv = iv * gv;
            const float hv = ov * fast_tanh(cv);
#pragma unroll
            for (int k = 0; k < 7; ++k) acc[r][k] = fmaf(hv, w[k], acc[r][k]);
        }
    }

    // ---- reduce across the 16 lanes of each half (masks < 16 never cross halves) ----
#pragma unroll
    for (int r = 0; r < 8; ++r)
#pragma unroll
        for (int k = 0; k < 7; ++k) {
            float v = acc[r][k];
            v += __shfl_xor(v, 1, 32);
            v += __shfl_xor(v, 2, 32);
            v += __shfl_xor(v, 4, 32);
            v += __shfl_xor(v, 8, 32);
            acc[r][k] = v;
        }

    // lanes 0-7 handle rows t0+0..7 (lower-half data), lanes 16-23 rows t0+8..15
    float lin[7];
#pragma unroll
    for (int k = 0; k < 7; ++k) lin[k] = 0.0f;
#pragma unroll
    for (int r = 0; r < 8; ++r)
#pragma unroll
        for (int k = 0; k < 7; ++k)
            lin[k] = (l15 == r) ? acc[r][k] : lin[k];

    if (l15 < 8) {
        const int t = t0 + half * 8 + l15;
#pragma unroll
        for (int k = 0; k < 7; ++k) lin[k] += bout[k];

        const float e    = fast_sigmoid(lin[0]);
        const float mux  = lin[1];
        const float muy  = lin[2];
        const float stdx = __expf(lin[3]);
        const float stdy = __expf(lin[4]);
        const float rho  = fast_tanh(lin[5]);

        const float z1 = zn[t * 2 + 0];
        const float z2 = zn[t * 2 + 1];
        const float u  = un[t];

        float q = 1.0f - rho * rho;
        q = q > 0.0f ? q : 0.0f;
        const float x = fmaf(stdx, z1, mux);
        const float y = muy + rho * stdy * z1 + stdy * sqrtf(q) * z2;

        out[t * 3 + 0] = (u < e) ? 1.0f : 0.0f;
        out[t * 3 + 1] = x;
        out[t * 3 + 2] = y;
    }
}

extern "C" void kernel_launch(void* const* d_in, const int* in_sizes, int n_in,
                              void* d_out, int out_size, void* d_ws, size_t ws_size,
                              hipStream_t stream) {
    (void)in_sizes; (void)n_in; (void)d_ws; (void)ws_size; (void)out_size;
    const float* seq  = (const float*)d_in[0];
    const float* Wih  = (const float*)d_in[1];
    // d_in[2] = W_hh : dead in the reference computation
    const float* bih  = (const float*)d_in[3];
    const float* bhh  = (const float*)d_in[4];
    const float* Wout = (const float*)d_in[5];
    const float* bout = (const float*)d_in[6];
    const float* zn   = (const float*)d_in[7];
    const float* un   = (const float*)d_in[8];
    float* out = (float*)d_out;

    // T=65536 rows -> 4096 16-row tiles, 8 waves (tiles) per 256-thread block
    dim3 grid(512), block(256);
    lstm_fused_wmma_kernel<<<grid, block, 0, stream>>>(seq, Wih, bih, bhh, Wout, bout, zn, un, out);
}